// HIRL_73151882985571
// MI455X (gfx1250) — compile-verified
//
#include <hip/hip_runtime.h>
#include <hip/hip_bf16.h>
#include <math.h>

typedef __attribute__((ext_vector_type(2))) float v2f;
typedef __attribute__((ext_vector_type(8))) float v8f;

#define BB 256
#define NNEG 1000
#define HH 3
#define DD 128
#define NEG_PER_TILE 5          // 5 negatives * 3 hierarchies = 15 rows of the 16-row A tile
#define TILES_PER_B (NNEG / NEG_PER_TILE)   // 200

// ---------------------------------------------------------------- reductions
__device__ __forceinline__ float block_sum_128(float v, float* sh4) {
    // 128 threads = 4 wave32; result broadcast to all threads
    #pragma unroll
    for (int o = 16; o > 0; o >>= 1) v += __shfl_down(v, o, 32);
    __syncthreads();                       // protect sh4 from previous use
    if ((threadIdx.x & 31) == 0) sh4[threadIdx.x >> 5] = v;
    __syncthreads();
    return sh4[0] + sh4[1] + sh4[2] + sh4[3];
}

// --------------------------------------------------------- kernel 1: setup
// Normalize se, compute positive-path loss and labels; block 0 also zeroes
// the 128-float zero-row used by k_neg's padding lanes (ws is poisoned).
__global__ __launch_bounds__(128) void k_setup(
    const float* __restrict__ se,
    const float* __restrict__ c0, const float* __restrict__ c1, const float* __restrict__ c2,
    const int* __restrict__ i0, const int* __restrict__ i1, const int* __restrict__ i2,
    const int* __restrict__ index,
    float* __restrict__ se_norm, float* __restrict__ pos_loss, int* __restrict__ pos_lab,
    float* __restrict__ zbuf)
{
    __shared__ float sh4[4];
    const int b = blockIdx.x;
    const int t = threadIdx.x;
    if (b == 0) zbuf[t] = 0.0f;            // zero row for padded WMMA lanes
    const int idx = index[b];
    float prodp = 1.0f;
    #pragma unroll
    for (int h = 0; h < HH; ++h) {
        float x = se[(size_t)b * (HH * DD) + h * DD + t];
        float ss = block_sum_128(x * x, sh4);
        float xn = x / fmaxf(sqrtf(ss), 1e-12f);
        se_norm[(size_t)b * (HH * DD) + h * DD + t] = xn;

        const int*   ii = (h == 0) ? i0 : (h == 1) ? i1 : i2;
        const float* cb = (h == 0) ? c0 : (h == 1) ? c1 : c2;
        int c = ii[idx];
        if (t == 0) pos_lab[b * HH + h] = c;
        float ce  = cb[(size_t)c * DD + t];
        float dot = block_sum_128(xn * ce, sh4);
        float css = block_sum_128(ce * ce, sh4);
        float sim = dot / fmaxf(sqrtf(css), 1e-12f);
        prodp *= (sim + 1.0f) * 0.5f;
    }
    if (t == 0) pos_loss[b] = -logf(prodp + 1e-6f);
}

// ------------------------------------------------ kernel 2: negatives (WMMA)
// One wave per (b, tile). A = 15 gathered centroid rows (5 negs x 3 hiers),
// B columns 0..2 = normalized se[b,h]. Padding lanes read a zero row, so the
// inner loop is pure load -> fmac(sumsq) -> wmma with no masking VALU.
__global__ __launch_bounds__(32) void k_neg(
    const float* __restrict__ c0, const float* __restrict__ c1, const float* __restrict__ c2,
    const int* __restrict__ negc,            // [B, NNEG, HH]
    const float* __restrict__ se_norm,       // [B, HH, DD]
    const int* __restrict__ pos_lab,         // [B, HH]
    const float* __restrict__ zbuf,          // [DD] zeros
    float* __restrict__ tilebuf)             // [B, TILES_PER_B, 2]
{
    const int bid  = blockIdx.x;
    const int b    = bid / TILES_PER_B;
    const int tile = bid - b * TILES_PER_B;
    const int lane = threadIdx.x;            // 0..31 (wave32)
    const int m    = lane & 15;              // A-matrix row this lane feeds
    const int half = lane >> 4;              // 0: K%4 in {0,1}; 1: K%4 in {2,3}

    // Row m -> (negative nloc = m/3, hierarchy h = m%3); row 15 is padding.
    const int  nloc   = m / 3;
    const int  h      = m - nloc * 3;
    const bool avalid = (m < 15);
    const int  hc     = avalid ? h : 0;
    const int  nn     = tile * NEG_PER_TILE + (avalid ? nloc : 0);
    const int  cidx   = negc[(size_t)b * (NNEG * HH) + (size_t)nn * HH + hc];
    const float* cb   = (hc == 0) ? c0 : (hc == 1) ? c1 : c2;
    const float2* __restrict__ crow =
        (const float2*)(avalid ? (cb + (size_t)cidx * DD) : zbuf);

    const bool bvalid = (m < HH);
    const float2* __restrict__ srow =
        (const float2*)(bvalid ? (se_norm + ((size_t)b * HH + m) * DD) : zbuf);

    v8f acc = {0.f, 0.f, 0.f, 0.f, 0.f, 0.f, 0.f, 0.f};
    float sumsq = 0.0f;
    // K = 128 as 32 chunks of K=4. Lane layout per ISA 7.12.2 (32-bit A 16x4):
    // lanes 0-15 carry K = 4c+0,4c+1 ; lanes 16-31 carry K = 4c+2,4c+3.
    #pragma unroll
    for (int c = 0; c < 32; ++c) {
        float2 av = crow[c * 2 + half];
        float2 bv = srow[c * 2 + half];
        sumsq += av.x * av.x + av.y * av.y;
        v2f A;  A.x = av.x;  A.y = av.y;
        v2f Bm; Bm.x = bv.x; Bm.y = bv.y;
        acc = __builtin_amdgcn_wmma_f32_16x16x4_f32(
            /*neg_a=*/false, A, /*neg_b=*/false, Bm,
            /*c_mod=*/(short)0, acc, /*reuse_a=*/false, /*reuse_b=*/false);
    }

    // Exchange D fragments + per-row-half sumsq through LDS.
    __shared__ float sacc[32][8];
    __shared__ float snorm[32];
    __shared__ float partial[NEG_PER_TILE][2];
    #pragma unroll
    for (int i = 0; i < 8; ++i) sacc[lane][i] = acc[i];
    snorm[lane] = sumsq;
    __syncthreads();

    if (lane < NEG_PER_TILE) {
        const int n = tile * NEG_PER_TILE + lane;
        float prod = 1.0f;
        int   diff = 0;
        #pragma unroll
        for (int hh = 0; hh < HH; ++hh) {
            const int mm = lane * 3 + hh;                 // 0..14
            // D layout (32-bit C/D 16x16): VGPR i holds rows i (lanes 0-15) and i+8 (lanes 16-31)
            float raw = (mm < 8) ? sacc[hh][mm] : sacc[16 + hh][mm - 8];
            float ss  = snorm[mm] + snorm[mm + 16];       // both K-halves of row mm
            float sim = raw / fmaxf(sqrtf(ss), 1e-12f);
            prod *= (sim + 1.0f) * 0.5f;
            int ci = negc[(size_t)b * (NNEG * HH) + (size_t)n * HH + hh];
            diff |= (ci != pos_lab[b * HH + hh]) ? 1 : 0;
        }
        float tn   = diff ? 1.0f : 0.0f;
        float loss = -logf(1.0f - prod + 1e-6f);
        partial[lane][0] = loss * tn;
        partial[lane][1] = tn;
    }
    __syncthreads();
    if (lane == 0) {
        float wl = 0.0f, ts = 0.0f;
        #pragma unroll
        for (int i = 0; i < NEG_PER_TILE; ++i) { wl += partial[i][0]; ts += partial[i][1]; }
        tilebuf[((size_t)b * TILES_PER_B + tile) * 2 + 0] = wl;
        tilebuf[((size_t)b * TILES_PER_B + tile) * 2 + 1] = ts;
    }
}

// --------------------------------------------------------- kernel 3: finish
__global__ __launch_bounds__(256) void k_final(
    const float* __restrict__ pos_loss, const float* __restrict__ tilebuf,
    float* __restrict__ out)
{
    __shared__ float sh[256];
    const int b = threadIdx.x;
    float wl = 0.0f, ts = 0.0f;
    for (int t = 0; t < TILES_PER_B; ++t) {
        wl += tilebuf[((size_t)b * TILES_PER_B + t) * 2 + 0];
        ts += tilebuf[((size_t)b * TILES_PER_B + t) * 2 + 1];
    }
    float negl = wl / (ts + 1e-6f);
    sh[b] = pos_loss[b] + negl;
    __syncthreads();
    for (int s = 128; s > 0; s >>= 1) {
        if (b < s) sh[b] += sh[b + s];
        __syncthreads();
    }
    if (b == 0) out[0] = sh[0] / (float)(2 * BB);
}

// ------------------------------------------------------------------ launch
extern "C" void kernel_launch(void* const* d_in, const int* in_sizes, int n_in,
                              void* d_out, int out_size, void* d_ws, size_t ws_size,
                              hipStream_t stream) {
    const float* se    = (const float*)d_in[0];
    const float* c0    = (const float*)d_in[1];
    const float* c1    = (const float*)d_in[2];
    const float* c2    = (const float*)d_in[3];
    const int*   i2c0  = (const int*)d_in[4];
    const int*   i2c1  = (const int*)d_in[5];
    const int*   i2c2  = (const int*)d_in[6];
    const int*   index = (const int*)d_in[7];
    const int*   negc  = (const int*)d_in[8];

    float* ws       = (float*)d_ws;
    float* se_norm  = ws;                                  // B*H*D = 98304 f
    float* pos_loss = se_norm + (size_t)BB * HH * DD;      // 256 f
    int*   pos_lab  = (int*)(pos_loss + BB);               // 768 i
    float* tilebuf  = (float*)(pos_lab + BB * HH);         // B*200*2 = 102400 f
    float* zbuf     = tilebuf + (size_t)BB * TILES_PER_B * 2; // 128 f of zeros

    k_setup<<<BB, 128, 0, stream>>>(se, c0, c1, c2, i2c0, i2c1, i2c2, index,
                                    se_norm, pos_loss, pos_lab, zbuf);
    k_neg<<<BB * TILES_PER_B, 32, 0, stream>>>(c0, c1, c2, negc, se_norm,
                                               pos_lab, zbuf, tilebuf);
    k_final<<<1, 256, 0, stream>>>(pos_loss, tilebuf, (float*)d_out);
}